// HypergraphLayer_32160715112910
// MI455X (gfx1250) — compile-verified
//
#include <hip/hip_runtime.h>
#include <hip/hip_bf16.h>

#define BB 4
#define NN 10000
#define DD 64
#define EE 100000
#define AA 6
#define OUTD 64
#define KK 128            // 2*DD
#define LN_EPS 1e-5f

typedef __attribute__((ext_vector_type(2))) float v2f;
typedef __attribute__((ext_vector_type(8))) float v8f;

// ---------------------------------------------------------------------------
// 0) zero the accumulator workspace (must re-zero every launch: graph replay)
// ---------------------------------------------------------------------------
__global__ void zero_kernel(float* __restrict__ p, int n) {
    int i = blockIdx.x * blockDim.x + threadIdx.x;
    if (i < n) p[i] = 0.0f;
}

// ---------------------------------------------------------------------------
// 1) per-edge all-but-one product messages + scatter-add into agg[B,N,D]
//    block = 256 threads = 4 edges x 64 d-channels; loop over b inside.
// ---------------------------------------------------------------------------
__global__ __launch_bounds__(256) void message_kernel(
    const float* __restrict__ nf,        // [B,N,D] (row 0 treated as zero)
    const float* __restrict__ rel_emb,   // [R,D]
    const float* __restrict__ pos_emb,   // [A+1,D]
    const int*   __restrict__ edge_list, // [E,A]
    const int*   __restrict__ rel,       // [E]
    float* __restrict__ agg)             // [B,N,D] accumulator (pre-zeroed)
{
    const int t  = threadIdx.x;
    const int eg = blockIdx.x * 4 + (t >> 6);   // edge index
    if (eg >= EE) return;
    const int d  = t & 63;

    int nodes[AA];
#pragma unroll
    for (int a = 0; a < AA; ++a) nodes[a] = edge_list[eg * AA + a];

    const int   r    = rel[eg];
    const float reld = rel_emb[r * DD + d];

    float pos[AA];
#pragma unroll
    for (int a = 0; a < AA; ++a) pos[a] = pos_emb[(a + 1) * DD + d];

    for (int b = 0; b < BB; ++b) {
        const size_t base = (size_t)b * NN * DD;
        float g[AA];
#pragma unroll
        for (int a = 0; a < AA; ++a) {
            const int n = nodes[a];
            // node 0: nf zeroed, pos_emb[0] == ones  ->  gathered == 1.0
            g[a] = (n == 0) ? 1.0f : (nf[base + (size_t)n * DD + d] + pos[a]);
        }
        // all-but-one product via prefix/suffix
        float pf[AA];
        float p = 1.0f;
#pragma unroll
        for (int a = 0; a < AA; ++a) { pf[a] = p; p *= g[a]; }
        float s = 1.0f;
#pragma unroll
        for (int a = AA - 1; a >= 0; --a) {
            const float msg = pf[a] * s * reld;
            s *= g[a];
            const int n = nodes[a];
            if (n != 0) {  // out[:,0,:] is forced to 0 anyway -> skip traffic
                unsafeAtomicAdd(&agg[base + (size_t)n * DD + d], msg);
            }
        }
    }
}

// ---------------------------------------------------------------------------
// 2) h = [agg | nf] @ W^T + b, then LayerNorm.  M=B*N=40000 (2500 strips of
//    16 rows), K=128, N=64.  One 128-thread block per strip; wave w owns the
//    16x16 output tile at columns [16w, 16w+16) using V_WMMA_F32_16X16X4_F32.
// ---------------------------------------------------------------------------
#define LDSTRIDE 130   // pad 128-float rows to break LDS bank conflicts
#define HSTRIDE  66    // padded row stride of the 16x64 h tile

__global__ __launch_bounds__(128) void gemm_ln_kernel(
    const float* __restrict__ nf,     // [B,N,D] flattened as [40000,64]
    const float* __restrict__ Wm,     // [OUT, 2D] row-major
    const float* __restrict__ bias,   // [OUT]
    const float* __restrict__ gamma,  // [OUT]
    const float* __restrict__ beta,   // [OUT]
    const float* __restrict__ agg,    // [B,N,D] flattened as [40000,64]
    float* __restrict__ out)          // [B,N,OUT]
{
    __shared__ float ldsX[16 * LDSTRIDE];
    __shared__ float hbuf[16 * HSTRIDE];
    __shared__ float psum[16][8];
    __shared__ float psq[16][8];
    __shared__ float mu_s[16];
    __shared__ float rstd_s[16];

    const int t    = threadIdx.x;          // 0..127
    const int row0 = blockIdx.x * 16;      // global row base in [0,40000)

    // ---- stage X strip [16 x 128] into LDS: col = t, loop rows -------------
    for (int r = 0; r < 16; ++r) {
        const int gr = row0 + r;
        const int n  = gr % NN;            // node index within batch
        float v;
        if (t < 64) {
            v = agg[(size_t)gr * 64 + t];              // agg row 0 already 0
        } else {
            v = (n == 0) ? 0.0f : nf[(size_t)gr * 64 + (t - 64)];
        }
        ldsX[r * LDSTRIDE + t] = v;
    }
    __syncthreads();

    // ---- WMMA: each wave computes a 16x16 tile over K=128 ------------------
    const int wave = t >> 5;
    const int lane = t & 31;
    const int m    = lane & 15;            // A row / D col-lane
    const int kb   = (lane >> 4) << 1;     // K sub-offset {0,2}
    const int ncol = wave * 16 + (lane & 15);  // global output column

    v8f c = {};
    for (int k = 0; k < KK; k += 4) {
        v2f a, bfrag;
        a.x = ldsX[m * LDSTRIDE + k + kb];
        a.y = ldsX[m * LDSTRIDE + k + kb + 1];
        // B = W^T : B[k][n] = W[n][k]
        bfrag.x = Wm[ncol * KK + k + kb];
        bfrag.y = Wm[ncol * KK + k + kb + 1];
        c = __builtin_amdgcn_wmma_f32_16x16x4_f32(
                /*neg_a=*/false, a, /*neg_b=*/false, bfrag,
                /*c_mod=*/(short)0, c, /*reuse_a=*/false, /*reuse_b=*/false);
    }

    // ---- tile -> LDS with bias (C/D layout: row = v + 8*(lane>=16)) --------
    {
        const float bv  = bias[ncol];
        const int   lhi = lane >> 4;
#pragma unroll
        for (int v = 0; v < 8; ++v) {
            const int mr = v + 8 * lhi;
            hbuf[mr * HSTRIDE + ncol] = c[v] + bv;
        }
    }
    __syncthreads();

    // ---- LayerNorm over 64 cols per row: 8 threads/row, 8 cols/thread ------
    {
        const int r = t >> 3;
        const int j = t & 7;
        float s = 0.0f, sq = 0.0f;
#pragma unroll
        for (int cix = j * 8; cix < j * 8 + 8; ++cix) {
            const float v = hbuf[r * HSTRIDE + cix];
            s += v; sq += v * v;
        }
        psum[r][j] = s;
        psq[r][j]  = sq;
    }
    __syncthreads();
    if (t < 16) {
        float s = 0.0f, sq = 0.0f;
#pragma unroll
        for (int j = 0; j < 8; ++j) { s += psum[t][j]; sq += psq[t][j]; }
        const float mu  = s * (1.0f / 64.0f);
        const float var = sq * (1.0f / 64.0f) - mu * mu;
        mu_s[t]   = mu;
        rstd_s[t] = rsqrtf(var + LN_EPS);
    }
    __syncthreads();
    {
        const int r  = t >> 3;
        const int j  = t & 7;
        const int gr = row0 + r;
        const float mu = mu_s[r];
        const float rs = rstd_s[r];
#pragma unroll
        for (int cix = j * 8; cix < j * 8 + 8; ++cix) {
            const float v = (hbuf[r * HSTRIDE + cix] - mu) * rs * gamma[cix] + beta[cix];
            out[(size_t)gr * 64 + cix] = v;
        }
    }
}

// ---------------------------------------------------------------------------
extern "C" void kernel_launch(void* const* d_in, const int* in_sizes, int n_in,
                              void* d_out, int out_size, void* d_ws, size_t ws_size,
                              hipStream_t stream) {
    const float* nf      = (const float*)d_in[0];
    // d_in[1] = query (unused by the reference)
    const float* rel_emb = (const float*)d_in[2];
    const float* pos_emb = (const float*)d_in[3];
    const float* Wm      = (const float*)d_in[4];
    const float* bias    = (const float*)d_in[5];
    const float* gamma   = (const float*)d_in[6];
    const float* beta    = (const float*)d_in[7];
    const int* edge_list = (const int*)d_in[8];
    const int* rel       = (const int*)d_in[9];
    float* out = (float*)d_out;
    float* agg = (float*)d_ws;   // B*N*D floats = 10.24 MB

    const int aggN = BB * NN * DD;
    zero_kernel<<<(aggN + 255) / 256, 256, 0, stream>>>(agg, aggN);
    message_kernel<<<(EE + 3) / 4, 256, 0, stream>>>(nf, rel_emb, pos_emb,
                                                     edge_list, rel, agg);
    gemm_ln_kernel<<<(BB * NN) / 16, 128, 0, stream>>>(nf, Wm, bias, gamma,
                                                       beta, agg, out);
}